// GNNWithFeatures_64776696758503
// MI455X (gfx1250) — compile-verified
//
#include <hip/hip_runtime.h>
#include <hip/hip_bf16.h>

#define N_NODES 50000
#define N_EDGES 800000
#define F_IN 128
#define HID 128
#define F_ADD 32
#define N_GRAPHS 128

typedef __attribute__((ext_vector_type(2))) float v2f;
typedef __attribute__((ext_vector_type(8))) float v8f;

// ---------------------------------------------------------------------------
// Generic fill
// ---------------------------------------------------------------------------
__global__ void fill_kernel(float* __restrict__ p, float v, long long n) {
    long long t = (long long)blockIdx.x * blockDim.x + threadIdx.x;
    if (t < n) p[t] = v;
}

// ---------------------------------------------------------------------------
// Degree accumulation over edges (deg pre-initialized to 1.0 for self-loops)
// ---------------------------------------------------------------------------
__global__ void deg_kernel(const long long* __restrict__ dst, float* __restrict__ deg,
                           int nEdges) {
    int e = blockIdx.x * blockDim.x + threadIdx.x;
    if (e < nEdges) atomicAdd(&deg[(int)dst[e]], 1.0f);
}

__global__ void dinv_kernel(const float* __restrict__ deg, float* __restrict__ dinv, int n) {
    int t = blockIdx.x * blockDim.x + threadIdx.x;
    if (t < n) dinv[t] = rsqrtf(deg[t]);   // deg >= 1 always (self loop)
}

// ---------------------------------------------------------------------------
// Y[n, 0:128] = X[n, 0:128] @ W[128,128] via V_WMMA_F32_16X16X4_F32.
// Block = 256 threads (8 waves). Each block does 128 rows; each wave does a
// 16-row stripe x all 128 cols (8 col-tiles of v8f accumulators).
//
// W is staged in LDS in a K-paired layout: sW[k/2][col][2] holds
// (W[2m][col], W[2m+1][col]) contiguously so each B-fragment is a single
// aligned ds_load_b64. A-fragment rows are clamped (not zeroed): an invalid
// A row only feeds an output row that is never stored.
// ---------------------------------------------------------------------------
__global__ __launch_bounds__(256) void gemm128_wmma(const float* __restrict__ X,
                                                    const float* __restrict__ W,
                                                    float* __restrict__ Y,
                                                    int nrows) {
    __shared__ float sW[128 * 128];   // [64 pair-rows][128 cols][2]
    for (int i = threadIdx.x; i < 128 * 128; i += 256) {
        int k = i >> 7, col = i & 127;
        sW[(k >> 1) * 256 + col * 2 + (k & 1)] = W[i];
    }
    __syncthreads();

    const int wave  = threadIdx.x >> 5;      // 0..7
    const int lane  = threadIdx.x & 31;
    const int lhalf = lane >> 4;             // 0: K+{0,1}, 1: K+{2,3}
    const int l15   = lane & 15;

    const int rowBase = blockIdx.x * 128 + wave * 16;
    const int arow    = rowBase + l15;
    const int crow    = arow < nrows ? arow : (nrows - 1);   // clamp, no branch
    const float* xrow = X + (size_t)crow * 128 + lhalf * 2;

    v8f acc[8] = {};

    for (int k = 0; k < 128; k += 4) {
        const v2f a = *(const v2f*)(xrow + k);                       // global_load_b64
        const float* wbase = sW + (((k >> 1) + lhalf) * 128 + l15) * 2;
#pragma unroll
        for (int c = 0; c < 8; ++c) {
            const v2f b = *(const v2f*)(wbase + c * 32);             // ds_load_b64
            acc[c] = __builtin_amdgcn_wmma_f32_16x16x4_f32(
                /*neg_a=*/false, a, /*neg_b=*/false, b,
                /*c_mod=*/(short)0, acc[c], /*reuse_a=*/false, /*reuse_b=*/false);
        }
    }

#pragma unroll
    for (int c = 0; c < 8; ++c) {
#pragma unroll
        for (int r = 0; r < 8; ++r) {
            const int row = rowBase + r + lhalf * 8;  // C/D layout: VGPR r -> M=r / M=r+8
            if (row < nrows) Y[(size_t)row * 128 + c * 16 + l15] = acc[c][r];
        }
    }
}

// ---------------------------------------------------------------------------
// Edge scatter: AGG[dst] += H[src] * (dinv[src]*dinv[dst]).
// One thread per (edge, 4 features): E*32 threads.
// ---------------------------------------------------------------------------
__global__ void scatter_kernel(const long long* __restrict__ src,
                               const long long* __restrict__ dst,
                               const float* __restrict__ dinv,
                               const float* __restrict__ H,
                               float* __restrict__ AGG, int nEdges) {
    long long t = (long long)blockIdx.x * blockDim.x + threadIdx.x;
    int e = (int)(t >> 5);
    if (e >= nEdges) return;
    int f = (int)(t & 31) * 4;
    int s = (int)src[e], d = (int)dst[e];
    float nrm = dinv[s] * dinv[d];
    const float* hp = H + (size_t)s * 128 + f;
    float* ap = AGG + (size_t)d * 128 + f;
    atomicAdd(ap + 0, hp[0] * nrm);
    atomicAdd(ap + 1, hp[1] * nrm);
    atomicAdd(ap + 2, hp[2] * nrm);
    atomicAdd(ap + 3, hp[3] * nrm);
}

// ---------------------------------------------------------------------------
// H[n,f] = relu(AGG[n,f] + HRAW[n,f]*dinv[n]^2 + bias[f])   (self-loop folded)
// ---------------------------------------------------------------------------
__global__ void bias_relu_kernel(const float* __restrict__ AGG,
                                 const float* __restrict__ HRAW,
                                 const float* __restrict__ dinv,
                                 const float* __restrict__ bias,
                                 float* __restrict__ H, long long total) {
    long long t = (long long)blockIdx.x * blockDim.x + threadIdx.x;
    if (t >= total) return;
    int n = (int)(t >> 7);
    int f = (int)(t & 127);
    float di = dinv[n];
    float v = AGG[t] + HRAW[t] * (di * di) + bias[f];
    H[t] = v > 0.0f ? v : 0.0f;
}

// ---------------------------------------------------------------------------
// Mean-pool accumulation
// ---------------------------------------------------------------------------
__global__ void pool_kernel(const float* __restrict__ H,
                            const long long* __restrict__ batch,
                            float* __restrict__ pooled,
                            float* __restrict__ counts, long long total) {
    long long t = (long long)blockIdx.x * blockDim.x + threadIdx.x;
    if (t >= total) return;
    int n = (int)(t >> 7);
    int f = (int)(t & 127);
    int g = (int)batch[n];
    atomicAdd(&pooled[(size_t)g * 128 + f], H[t]);
    if (f == 0) atomicAdd(&counts[g], 1.0f);
}

// ---------------------------------------------------------------------------
// Per-graph head: z = [pooled/cnt, add_feat] (160) -> fc1(160,128)+relu -> fc2(128,2)
// One block per graph, 128 threads.
// ---------------------------------------------------------------------------
__global__ __launch_bounds__(128) void head_kernel(const float* __restrict__ pooled,
                                                   const float* __restrict__ counts,
                                                   const float* __restrict__ addf,
                                                   const float* __restrict__ fc1_w,
                                                   const float* __restrict__ fc1_b,
                                                   const float* __restrict__ fc2_w,
                                                   const float* __restrict__ fc2_b,
                                                   float* __restrict__ out) {
    __shared__ float z[HID + F_ADD];
    __shared__ float hid[HID];
    const int g = blockIdx.x;
    const int t = threadIdx.x;

    float cnt = counts[g];
    cnt = cnt > 1.0f ? cnt : 1.0f;
    z[t] = pooled[(size_t)g * 128 + t] / cnt;
    if (t < F_ADD) z[HID + t] = addf[(size_t)g * F_ADD + t];
    __syncthreads();

    float acc = fc1_b[t];
    for (int k = 0; k < HID + F_ADD; ++k) acc += z[k] * fc1_w[(size_t)k * HID + t];
    hid[t] = acc > 0.0f ? acc : 0.0f;
    __syncthreads();

    if (t < 2) {
        float o = fc2_b[t];
        for (int k = 0; k < HID; ++k) o += hid[k] * fc2_w[(size_t)k * 2 + t];
        out[(size_t)g * 2 + t] = o;
    }
}

// ---------------------------------------------------------------------------
extern "C" void kernel_launch(void* const* d_in, const int* in_sizes, int n_in,
                              void* d_out, int out_size, void* d_ws, size_t ws_size,
                              hipStream_t stream) {
    const float*     x     = (const float*)d_in[0];
    const long long* edges = (const long long*)d_in[1];   // [2, E] int64
    const long long* batch = (const long long*)d_in[2];   // [N] int64
    const float*     addf  = (const float*)d_in[3];
    const float*     W1    = (const float*)d_in[4];
    const float*     b1    = (const float*)d_in[5];
    const float*     W2    = (const float*)d_in[6];
    const float*     b2    = (const float*)d_in[7];
    const float*     fc1_w = (const float*)d_in[8];
    const float*     fc1_b = (const float*)d_in[9];
    const float*     fc2_w = (const float*)d_in[10];
    const float*     fc2_b = (const float*)d_in[11];
    float* out = (float*)d_out;

    const long long* src = edges;
    const long long* dst = edges + N_EDGES;

    // --- workspace carve-up ---
    size_t off = 0;
    auto carve = [&](size_t bytes) -> float* {
        float* p = (float*)((char*)d_ws + off);
        off += (bytes + 255) & ~(size_t)255;
        return p;
    };
    float* deg    = carve((size_t)N_NODES * 4);
    float* dinv   = carve((size_t)N_NODES * 4);
    float* hraw   = carve((size_t)N_NODES * 128 * 4);
    float* agg    = carve((size_t)N_NODES * 128 * 4);
    float* h      = carve((size_t)N_NODES * 128 * 4);
    float* pooled = carve((size_t)N_GRAPHS * 128 * 4);
    float* counts = carve((size_t)N_GRAPHS * 4);
    (void)ws_size; (void)n_in; (void)in_sizes; (void)out_size;

    const long long NF = (long long)N_NODES * 128;
    const int B = 256;
    const int gridNF    = (int)((NF + B - 1) / B);
    const int gridNodes = (N_NODES + B - 1) / B;
    const int gridEdges = (N_EDGES + B - 1) / B;
    const int gridScat  = (int)(((long long)N_EDGES * 32 + B - 1) / B);
    const int gridGemm  = (N_NODES + 127) / 128;

    // degree / normalization
    fill_kernel<<<gridNodes, B, 0, stream>>>(deg, 1.0f, N_NODES);   // self loops
    deg_kernel<<<gridEdges, B, 0, stream>>>(dst, deg, N_EDGES);
    dinv_kernel<<<gridNodes, B, 0, stream>>>(deg, dinv, N_NODES);

    // ---- layer 1 ----
    gemm128_wmma<<<gridGemm, B, 0, stream>>>(x, W1, hraw, N_NODES);
    fill_kernel<<<gridNF, B, 0, stream>>>(agg, 0.0f, NF);
    scatter_kernel<<<gridScat, B, 0, stream>>>(src, dst, dinv, hraw, agg, N_EDGES);
    bias_relu_kernel<<<gridNF, B, 0, stream>>>(agg, hraw, dinv, b1, h, NF);

    // ---- layer 2 ----
    gemm128_wmma<<<gridGemm, B, 0, stream>>>(h, W2, hraw, N_NODES);
    fill_kernel<<<gridNF, B, 0, stream>>>(agg, 0.0f, NF);
    scatter_kernel<<<gridScat, B, 0, stream>>>(src, dst, dinv, hraw, agg, N_EDGES);
    bias_relu_kernel<<<gridNF, B, 0, stream>>>(agg, hraw, dinv, b2, h, NF);

    // ---- pooling ----
    fill_kernel<<<(N_GRAPHS * 128 + B - 1) / B, B, 0, stream>>>(pooled, 0.0f, N_GRAPHS * 128);
    fill_kernel<<<1, B, 0, stream>>>(counts, 0.0f, N_GRAPHS);
    pool_kernel<<<gridNF, B, 0, stream>>>(h, batch, pooled, counts, NF);

    // ---- head ----
    head_kernel<<<N_GRAPHS, 128, 0, stream>>>(pooled, counts, addf,
                                              fc1_w, fc1_b, fc2_w, fc2_b, out);
}